// NoRA_16346645528617
// MI455X (gfx1250) — compile-verified
//
#include <hip/hip_runtime.h>

typedef __attribute__((ext_vector_type(2))) float v2f;
typedef __attribute__((ext_vector_type(4))) float f4;
typedef __attribute__((ext_vector_type(8))) float v8f;

#define NORA_SCALING 2.0f   // lora_alpha / r = 6/3
#define NORA_EPS     1e-6f

// A-matrix (16x4, M x K): rows 0-7 = lora_B_numerator[8x3], rows 8-15 = lora_B_denominator[8x3], K=3 padded 0
__device__ __forceinline__ float nora_ldA(int m, int k, const float* lBn, const float* lBd) {
  if (k >= 3) return 0.0f;
  return (m < 8) ? lBn[m * 3 + k] : lBd[(m - 8) * 3 + k];
}
// B-matrix (4x16, K x N): cols 0-5 = lora_A_numerator[3x6], cols 6-9 = lora_A_denominator[3x4], rest 0
__device__ __forceinline__ float nora_ldB(int k, int n, const float* lAn, const float* lAd) {
  if (k >= 3) return 0.0f;
  if (n < 6)  return lAn[k * 6 + n];
  if (n < 10) return lAd[k * 4 + (n - 6)];
  return 0.0f;
}

__device__ __forceinline__ f4 nora_eval4(f4 v,
                                         float a0, float a1, float a2, float a3, float a4, float a5,
                                         float b0, float b1, float b2, float b3) {
  f4 r;
  #pragma unroll
  for (int i = 0; i < 4; ++i) {
    const float t = v[i];
    float p = __builtin_fmaf(a5, t, a4);          // Horner P5
    p = __builtin_fmaf(p, t, a3);
    p = __builtin_fmaf(p, t, a2);
    p = __builtin_fmaf(p, t, a1);
    p = __builtin_fmaf(p, t, a0);
    float z = __builtin_fmaf(b3, t, b2);          // Horner P3
    z = __builtin_fmaf(z, t, b1);
    z = __builtin_fmaf(z, t, b0);
    const float q = 1.0f + __builtin_fabsf(z * t);   // q >= 1: rcp is safe
    float rq = __builtin_amdgcn_rcpf(q);             // v_rcp_f32 (TRANS, co-executes)
    const float e = __builtin_fmaf(-q, rq, 1.0f);    // one Newton step -> ~1 ulp
    rq = __builtin_fmaf(rq, e, rq);
    r[i] = p * rq;
  }
  return r;
}

__global__ __launch_bounds__(256)
void nora_rational_kernel(const float* __restrict__ x,
                          const float* __restrict__ bn,
                          const float* __restrict__ bd,
                          const float* __restrict__ lAn,
                          const float* __restrict__ lBn,
                          const float* __restrict__ lAd,
                          const float* __restrict__ lBd,
                          float* __restrict__ out,
                          int rows)   // B*L = 16384
{
  __shared__ float s_num[8][6];
  __shared__ float s_den[8][4];

  const int tid = threadIdx.x;

  // ---- One wave fuses both LoRA coefficient matmuls into a single f32 WMMA ----
  if (tid < 32) {
    const int lane  = tid;
    const int m     = lane & 15;              // M (A) / N (B) position
    const int kbase = (lane < 16) ? 0 : 2;    // K pair held by this half-wave

    v2f A, B;
    A.x = nora_ldA(m, kbase,     lBn, lBd);   // VGPR0: K=kbase
    A.y = nora_ldA(m, kbase + 1, lBn, lBd);   // VGPR1: K=kbase+1
    B.x = nora_ldB(kbase,     m, lAn, lAd);
    B.y = nora_ldB(kbase + 1, m, lAn, lAd);

    v8f C = {};
    C = __builtin_amdgcn_wmma_f32_16x16x4_f32(
        /*neg_a=*/false, A, /*neg_b=*/false, B,
        /*c_mod=*/(short)0, C, /*reuse_a=*/false, /*reuse_b=*/false);

    // D layout: VGPR j -> rows j (lanes 0-15, N=lane) and j+8 (lanes 16-31, N=lane-16)
    if (lane < 6) {                            // numerator update: D[g][lane], g=0..7
      #pragma unroll
      for (int g = 0; g < 8; ++g)
        s_num[g][lane] = bn[g * 6 + lane] + NORA_SCALING * C[g];
    } else if (lane >= 22 && lane < 26) {      // denominator update: D[8+g][6+kk] -> lane = 22+kk
      const int kk = lane - 22;
      #pragma unroll
      for (int g = 0; g < 8; ++g)
        s_den[g][kk] = bd[g * 4 + kk] + NORA_SCALING * C[g] + NORA_EPS;
    }
  }
  __syncthreads();

  // ---- Streaming rational evaluation (memory-bound: b128 NT loads/stores) ----
  // Block owns a fixed 1024-channel chunk -> each thread's group is constant.
  const int cbase = blockIdx.x * 1024 + tid * 4;   // first channel of this thread's float4
  const int g     = cbase >> 9;                    // 512 channels per group

  const float a0 = s_num[g][0], a1 = s_num[g][1], a2 = s_num[g][2];
  const float a3 = s_num[g][3], a4 = s_num[g][4], a5 = s_num[g][5];
  const float b0 = s_den[g][0], b1 = s_den[g][1];
  const float b2 = s_den[g][2], b3 = s_den[g][3];

  const int rstride = gridDim.y;
  for (int row = blockIdx.y; row < rows; row += 2 * rstride) {
    const int row2 = row + rstride;
    const long long idx0 = (long long)row * 4096 + cbase;

    // two independent b128 NT loads in flight per iteration (MLP)
    f4 v0 = __builtin_nontemporal_load((const f4*)(x + idx0));
    f4 v1;
    long long idx1 = 0;
    const bool have2 = (row2 < rows);
    if (have2) {
      idx1 = (long long)row2 * 4096 + cbase;
      v1 = __builtin_nontemporal_load((const f4*)(x + idx1));
    }

    f4 r0 = nora_eval4(v0, a0, a1, a2, a3, a4, a5, b0, b1, b2, b3);
    __builtin_nontemporal_store(r0, (f4*)(out + idx0));

    if (have2) {
      f4 r1 = nora_eval4(v1, a0, a1, a2, a3, a4, a5, b0, b1, b2, b3);
      __builtin_nontemporal_store(r1, (f4*)(out + idx1));
    }
  }
}

extern "C" void kernel_launch(void* const* d_in, const int* in_sizes, int n_in,
                              void* d_out, int out_size, void* d_ws, size_t ws_size,
                              hipStream_t stream) {
  const float* x   = (const float*)d_in[0];
  const float* bn  = (const float*)d_in[1];
  const float* bd  = (const float*)d_in[2];
  const float* lAn = (const float*)d_in[3];
  const float* lBn = (const float*)d_in[4];
  const float* lAd = (const float*)d_in[5];
  const float* lBd = (const float*)d_in[6];
  float* out = (float*)d_out;

  const int rows = in_sizes[0] / 4096;   // B*L = 16384

  // grid.x = 4 channel chunks of 1024; grid.y row-stride blocks (2 rows/iter, 4 iters/block)
  dim3 grid(4, 2048);
  nora_rational_kernel<<<grid, 256, 0, stream>>>(x, bn, bd, lAn, lBn, lAd, lBd, out, rows);
}